// CTProjector2DModule_32306744000769
// MI455X (gfx1250) — compile-verified
//
#include <hip/hip_runtime.h>
#include <hip/hip_bf16.h>

typedef float v2f __attribute__((ext_vector_type(2)));
typedef float v8f __attribute__((ext_vector_type(8)));

#define N_ROWI 256
#define N_COLI 256
#define N_BAT  4

// ds_swizzle_b32 XOR pattern within group-of-32: offset = {0, xor[4:0], or=0, and=0x1f}
#define SWIZ_XOR(x, m) \
    __int_as_float(__builtin_amdgcn_ds_swizzle(__float_as_int(x), ((m) << 10) | 0x1f))

// Repack image (4,256,256) -> (256*256) float4 so one b128 load serves all 4 batches.
__global__ void ct_repack_image(const float* __restrict__ img, float4* __restrict__ packed) {
    int idx = blockIdx.x * blockDim.x + threadIdx.x;          // 65536 threads total
    const int plane = N_ROWI * N_COLI;
    float4 v;
    v.x = img[idx];
    v.y = img[idx + plane];
    v.z = img[idx + 2 * plane];
    v.w = img[idx + 3 * plane];
    packed[idx] = v;
}

template <bool PACKED>
__global__ __launch_bounds__(256)
void ct_siddon_fwd(const float*  __restrict__ img,     // (4,256,256) if !PACKED
                   const float4* __restrict__ imgp,    // (65536) if PACKED
                   const float*  __restrict__ tvals,   // (n_ray, n_t) sorted
                   const float*  __restrict__ Mmat,    // 4
                   const float*  __restrict__ bvec,    // 2
                   const float*  __restrict__ src,     // (n_ray,2)
                   const float*  __restrict__ dst,     // (n_ray,2)
                   float*        __restrict__ out,     // (4, n_ray)
                   int n_ray, int n_t) {
    const int lane = threadIdx.x & 31;
    const int wib  = threadIdx.x >> 5;
    const int ray  = blockIdx.x * (blockDim.x >> 5) + wib;
    if (ray >= n_ray) return;          // wave-uniform: EXEC stays all-ones

    // ---- per-ray (wave-uniform) setup: p0 = Minv@(src-b), d = Minv@(dst-src) ----
    const float m00 = Mmat[0], m01 = Mmat[1], m10 = Mmat[2], m11 = Mmat[3];
    const float det = m00 * m11 - m01 * m10;
    const float i00 =  m11 / det, i01 = -m01 / det;
    const float i10 = -m10 / det, i11 =  m00 / det;
    const float bx = bvec[0], by = bvec[1];
    const float sx = src[2 * ray], sy = src[2 * ray + 1];
    const float ex = dst[2 * ray], ey = dst[2 * ray + 1];
    // +0.5 pre-bias: r = floor(q0 + tm*d) == floor(p0 + tm*d + 0.5)
    const float q0x = i00 * (sx - bx) + i01 * (sy - by) + 0.5f;
    const float q0y = i10 * (sx - bx) + i11 * (sy - by) + 0.5f;
    const float rdx = ex - sx, rdy = ey - sy;
    const float dx = i00 * rdx + i01 * rdy;
    const float dy = i10 * rdx + i11 * rdy;
    const float ray_len = sqrtf(rdx * rdx + rdy * rdy);
    const float* __restrict__ row = tvals + (size_t)ray * (size_t)n_t;

    const int n_seg = n_t - 1;                 // 513
    const int L     = n_t - 1;                 // last valid t index (513)
    const int plane = N_ROWI * N_COLI;
    float a0 = 0.f, a1 = 0.f, a2 = 0.f, a3 = 0.f;

    // one Siddon segment: weight * gathered 4-batch pixel
    auto proc = [&](float t0, float t1) {
        float seg = (t1 - t0) * ray_len;
        float tm  = 0.5f * (t0 + t1);
        float ux  = q0x + tm * dx;
        float uy  = q0y + tm * dy;
        int r = (int)floorf(ux);
        int c = (int)floorf(uy);
        bool valid = (r >= 0) & (r < N_ROWI) & (c >= 0) & (c < N_COLI) & (seg > 0.0f);
        int rc = min(max(r, 0), N_ROWI - 1);
        int cc = min(max(c, 0), N_COLI - 1);
        int idx = rc * N_COLI + cc;
        float w = valid ? seg : 0.0f;
        if (PACKED) {
            float4 v = imgp[idx];            // single b128 gather, L2-resident
            a0 = fmaf(w, v.x, a0);
            a1 = fmaf(w, v.y, a1);
            a2 = fmaf(w, v.z, a2);
            a3 = fmaf(w, v.w, a3);
        } else {
            a0 = fmaf(w, img[idx],             a0);
            a1 = fmaf(w, img[idx + plane],     a1);
            a2 = fmaf(w, img[idx + 2 * plane], a2);
            a3 = fmaf(w, img[idx + 3 * plane], a3);
        }
    };

    // ---- main loop: 2 segments per lane, 64 per wave-iter; all reads provably
    //      in-row (block b needs t up to (b+1)*64 <= n_seg), so no masking. ----
    const int nfull = n_seg >> 6;              // 8 full blocks for n_seg=513
    for (int it = 0; it < nfull; ++it) {
        int s0 = (it << 6) + (lane << 1);
        float2 tp = *reinterpret_cast<const float2*>(row + s0);  // 8B-aligned always
        float t2  = row[s0 + 2];
        if (it + 2 <= nfull)
            __builtin_prefetch(row + s0 + 128, 0, 0);            // global_prefetch_b8
        proc(tp.x, tp.y);
        proc(tp.y, t2);
    }

    // ---- remainder (s = nfull*64 .. n_seg-1): clamped duplicate-read trick,
    //      clamped lanes get t0==t1 -> seg==0 -> rejected by seg>0. Uniform trips. ----
    const int rem_base  = nfull << 6;
    const int rem_iters = (n_seg - rem_base + 31) >> 5;          // 1 for n_seg=513
    for (int k = 0; k < rem_iters; ++k) {
        int s  = rem_base + (k << 5) + lane;
        int i0 = min(s, L);
        int i1 = min(s + 1, L);
        proc(row[i0], row[i1]);
    }

    // ---- cross-lane reduction, stage 1 via V_WMMA_F32_16X16X4_F32 (exact f32) ----
    // B carries accumulators: lane n (<16): {B[0][n],B[1][n]}, lane 16+n: {B[2][n],B[3][n]}.
    // A row0 = (1,0,1,0) -> C[0][n] = acc(lane n) + acc(lane n+16); row1 = (0,1,0,1).
    // Two WMMAs fold all 4 accumulators into C rows 0..3 (rows 4..15 stay zero).
    int l = lane & 15;
    v2f A1 = {0.0f, 0.0f};
    v2f A2 = {0.0f, 0.0f};
    if (l == 0) A1.x = 1.0f;   // lanes 0,16 -> A[0][0]=A[0][2]=1
    if (l == 1) A1.y = 1.0f;   // lanes 1,17 -> A[1][1]=A[1][3]=1
    if (l == 2) A2.x = 1.0f;   // lanes 2,18 -> A[2][0]=A[2][2]=1
    if (l == 3) A2.y = 1.0f;   // lanes 3,19 -> A[3][1]=A[3][3]=1
    v2f B1; B1.x = a0; B1.y = a1;
    v2f B2; B2.x = a2; B2.y = a3;
    v8f C = {};
    C = __builtin_amdgcn_wmma_f32_16x16x4_f32(false, A1, false, B1, (short)0, C, false, false);
    C = __builtin_amdgcn_wmma_f32_16x16x4_f32(false, A2, false, B2, (short)0, C, false, false);

    // C rows 0..3 (VGPR 0..3, lanes 0..15) hold half-folded partials; lanes 16..31
    // hold rows 8..15 == 0, so XOR swizzles with masks 8,4,2,1 (within 16-lane
    // halves) finish the column sum without bpermute address setup.
    float s0 = C[0], s1 = C[1], s2 = C[2], s3 = C[3];
    s0 += SWIZ_XOR(s0, 8);  s1 += SWIZ_XOR(s1, 8);
    s2 += SWIZ_XOR(s2, 8);  s3 += SWIZ_XOR(s3, 8);
    s0 += SWIZ_XOR(s0, 4);  s1 += SWIZ_XOR(s1, 4);
    s2 += SWIZ_XOR(s2, 4);  s3 += SWIZ_XOR(s3, 4);
    s0 += SWIZ_XOR(s0, 2);  s1 += SWIZ_XOR(s1, 2);
    s2 += SWIZ_XOR(s2, 2);  s3 += SWIZ_XOR(s3, 2);
    s0 += SWIZ_XOR(s0, 1);  s1 += SWIZ_XOR(s1, 1);
    s2 += SWIZ_XOR(s2, 1);  s3 += SWIZ_XOR(s3, 1);

    if (lane == 0) {
        out[ray]             = s0;
        out[n_ray + ray]     = s1;
        out[2 * n_ray + ray] = s2;
        out[3 * n_ray + ray] = s3;
    }
}

extern "C" void kernel_launch(void* const* d_in, const int* in_sizes, int n_in,
                              void* d_out, int out_size, void* d_ws, size_t ws_size,
                              hipStream_t stream) {
    const float* img   = (const float*)d_in[0];   // (4,256,256)
    const float* tvals = (const float*)d_in[1];   // (n_ray, n_t)
    const float* M     = (const float*)d_in[2];   // (2,2)
    const float* b     = (const float*)d_in[3];   // (2,)
    const float* src   = (const float*)d_in[4];   // (n_ray,2)
    const float* dst   = (const float*)d_in[5];   // (n_ray,2)
    float* out = (float*)d_out;

    const int n_ray = in_sizes[4] / 2;
    const int n_t   = in_sizes[1] / n_ray;

    const int waves_per_block = 8;                 // 256 threads = 8 wave32
    const int block = waves_per_block * 32;
    const int grid  = (n_ray + waves_per_block - 1) / waves_per_block;

    const size_t packed_bytes = (size_t)N_ROWI * N_COLI * sizeof(float4);
    const bool use_packed = (ws_size >= packed_bytes) &&
                            (in_sizes[0] == N_BAT * N_ROWI * N_COLI);

    if (use_packed) {
        ct_repack_image<<<(N_ROWI * N_COLI) / 256, 256, 0, stream>>>(img, (float4*)d_ws);
        ct_siddon_fwd<true><<<grid, block, 0, stream>>>(
            img, (const float4*)d_ws, tvals, M, b, src, dst, out, n_ray, n_t);
    } else {
        ct_siddon_fwd<false><<<grid, block, 0, stream>>>(
            img, nullptr, tvals, M, b, src, dst, out, n_ray, n_t);
    }
}